// Block_69191923139027
// MI455X (gfx1250) — compile-verified
//
#include <hip/hip_runtime.h>
#include <hip/hip_bf16.h>

// ---------------- problem constants (match reference) ----------------
#define Bq   4
#define Tq   2048
#define SDq  64
#define NEq  4096
#define Hq   8
#define HSq  8
#define FHq  256
#define EPSq 1e-5f

typedef __attribute__((ext_vector_type(2))) float v2f;
typedef __attribute__((ext_vector_type(8))) float v8f;

static __device__ __forceinline__ v8f wmma4(v2f a, v2f b, v8f c) {
    // V_WMMA_F32_16X16X4_F32 : D = A(16x4) * B(4x16) + C(16x16), fp32
    return __builtin_amdgcn_wmma_f32_16x16x4_f32(
        /*neg_a=*/false, a, /*neg_b=*/false, b,
        /*c_mod=*/(short)0, c, /*reuse_a=*/false, /*reuse_b=*/false);
}

// =====================================================================
// K1: Feebler  h0[b,i,t] = sum_j x[b,i,j,t] * fw[i,j,t]
// flat layout of h0 (B,SD,T) == (B,T,SD) reinterpretation used downstream
// =====================================================================
__global__ __launch_bounds__(256) void k_feebler(
    const float* __restrict__ x, const float* __restrict__ fw,
    float* __restrict__ h0)
{
    unsigned id = blockIdx.x * 256u + threadIdx.x;        // < B*SD*T = 524288
    int t = id & (Tq - 1);
    unsigned r = id >> 11;
    int i = r & (SDq - 1);
    int b = r >> 6;
    const float* xp = x  + (((size_t)(b * SDq + i) * SDq) * Tq + t);
    const float* wp = fw + (((size_t)(i * SDq)) * Tq + t);
    float acc = 0.f;
#pragma unroll 8
    for (int j = 0; j < SDq; ++j)
        acc += xp[(size_t)j * Tq] * wp[(size_t)j * Tq];
    h0[id] = acc;
}

// =====================================================================
// K2: LN1 + q/k/v projections (one wave per token row) + ck/cv reduction
// =====================================================================
__global__ __launch_bounds__(256) void k_ln1_qkv(
    const float* __restrict__ h0,
    const float* __restrict__ g1, const float* __restrict__ b1v,
    const float* __restrict__ wq, const float* __restrict__ wk,
    const float* __restrict__ wvw,
    float* __restrict__ qout, float* __restrict__ ck, float* __restrict__ cv)
{
    __shared__ float ylds[8][64];
    __shared__ float kred[64], vred[64];
    const int tid  = threadIdx.x;
    const int lane = tid & 31;
    const int wid  = tid >> 5;
    const int row  = blockIdx.x * 8 + wid;                // [0, 8192)
    const int b    = row >> 11;

    if (tid < 64) { kred[tid] = 0.f; vred[tid] = 0.f; }

    // ---- LayerNorm over 64 (each lane owns cols lane, lane+32)
    float e0 = h0[row * SDq + lane];
    float e1 = h0[row * SDq + 32 + lane];
    float s  = e0 + e1, s2 = e0 * e0 + e1 * e1;
#pragma unroll
    for (int off = 16; off >= 1; off >>= 1) {
        s  += __shfl_xor(s,  off);
        s2 += __shfl_xor(s2, off);
    }
    float mean = s * (1.f / 64.f);
    float var  = s2 * (1.f / 64.f) - mean * mean;
    float rstd = rsqrtf(var + EPSq);
    ylds[wid][lane]      = (e0 - mean) * rstd * g1[lane]      + b1v[lane];
    ylds[wid][lane + 32] = (e1 - mean) * rstd * g1[lane + 32] + b1v[lane + 32];
    __syncthreads();

    // ---- q/k/v : each lane produces 2 output columns of each
    const int c0 = lane, c1 = lane + 32;
    const int hA = c0 >> 3, dA = c0 & 7;
    const int hB = c1 >> 3, dB = c1 & 7;
    float qa = 0, qb = 0, ka = 0, kb = 0, va = 0, vb = 0;
#pragma unroll 4
    for (int sI = 0; sI < SDq; ++sI) {
        float ys = ylds[wid][sI];
        int o0 = hA * (SDq * HSq) + sI * HSq + dA;
        int o1 = hB * (SDq * HSq) + sI * HSq + dB;
        qa += ys * wq[o0];  qb += ys * wq[o1];
        ka += ys * wk[o0];  kb += ys * wk[o1];
        va += ys * wvw[o0]; vb += ys * wvw[o1];
    }
    qout[row * SDq + c0] = qa;
    qout[row * SDq + c1] = qb;
    atomicAdd(&kred[c0], ka); atomicAdd(&kred[c1], kb);
    atomicAdd(&vred[c0], va); atomicAdd(&vred[c1], vb);
    __syncthreads();
    if (tid < 64) {
        atomicAdd(&ck[b * SDq + tid], kred[tid]);
        atomicAdd(&cv[b * SDq + tid], vred[tid]);
    }
}

// =====================================================================
// K3: softmax statistics over T per (b, col):  m = max_t z, l = sum exp(z-m)
//     z[t] = q[b,t,c] * ck[b,c]
// =====================================================================
__global__ __launch_bounds__(256) void k_softmax_stats(
    const float* __restrict__ q, const float* __restrict__ ck,
    float* __restrict__ mout, float* __restrict__ lout)
{
    __shared__ float red[256];
    const int tid = threadIdx.x;
    const int idx = blockIdx.x;            // b*64 + c
    const int b   = idx >> 6;
    const int c   = idx & 63;
    const float ckv = ck[idx];

    float z[8];
    float mx = -3.4e38f;
#pragma unroll
    for (int k = 0; k < 8; ++k) {
        int t = tid + k * 256;
        z[k] = q[((size_t)(b * Tq + t)) * SDq + c] * ckv;
        mx = fmaxf(mx, z[k]);
    }
    red[tid] = mx; __syncthreads();
#pragma unroll
    for (int sft = 128; sft >= 1; sft >>= 1) {
        if (tid < sft) red[tid] = fmaxf(red[tid], red[tid + sft]);
        __syncthreads();
    }
    float m0 = red[0];
    __syncthreads();
    float se = 0.f;
#pragma unroll
    for (int k = 0; k < 8; ++k) se += __expf(z[k] - m0);
    red[tid] = se; __syncthreads();
#pragma unroll
    for (int sft = 128; sft >= 1; sft >>= 1) {
        if (tid < sft) red[tid] += red[tid + sft];
        __syncthreads();
    }
    if (tid == 0) { mout[idx] = m0; lout[idx] = red[0]; }
}

// =====================================================================
// K4: fused  sa = cv*softmax(q*ck) ; h1 = h0 + sa@proj + pb ;
//            y2 = LN2(h1) ; h = h1 + relu(y2@w1+b1)@w2+b2
// One wave (32 threads) per 16-row tile; all GEMMs via WMMA f32 16x16x4.
// =====================================================================
__global__ __launch_bounds__(32) void k_attn_ffn(
    const float* __restrict__ q,  const float* __restrict__ h0,
    const float* __restrict__ ck, const float* __restrict__ cv,
    const float* __restrict__ mM, const float* __restrict__ lL,
    const float* __restrict__ pw, const float* __restrict__ pb,
    const float* __restrict__ g2, const float* __restrict__ b2v,
    const float* __restrict__ w1, const float* __restrict__ bb1,
    const float* __restrict__ w2, const float* __restrict__ bb2,
    float* __restrict__ hout)
{
    __shared__ float Abuf[16 * 65];     // A-tile stage (SA, then Y2); pitch 65
    __shared__ float H1[16 * 65];       // post-attention residual tile
    __shared__ float Hid[16 * 257];     // FFN hidden 16x256; pitch 257

    const int lane = threadIdx.x;
    const int lo = lane & 15;           // WMMA: A row / B,C,D column
    const int hi = lane >> 4;           // WMMA: K/M-half selector
    const int row0 = blockIdx.x * 16;   // 16 token rows per tile
    const int b = row0 >> 11;
    const float* ckb = ck + b * SDq;
    const float* cvb = cv + b * SDq;
    const float* mb  = mM + b * SDq;
    const float* lb  = lL + b * SDq;

    // ---- phase 1: sa tile = cv * softmax(q*ck)  (lane owns row lo, half hi)
#pragma unroll 4
    for (int c = 0; c < 32; ++c) {
        int cc = hi * 32 + c;
        float qv = q[((size_t)(row0 + lo)) * SDq + cc];
        float aw = __expf(qv * ckb[cc] - mb[cc]) / lb[cc];
        Abuf[lo * 65 + cc] = cvb[cc] * aw;
    }
    __syncthreads();

    // ---- phase 2: H1 = h0 + SA @ proj_w + proj_b   (WMMA)
    for (int n0 = 0; n0 < SDq; n0 += 16) {
        float bias = pb[n0 + lo];
        v8f acc;
#pragma unroll
        for (int j = 0; j < 8; ++j) acc[j] = bias;
#pragma unroll
        for (int k0 = 0; k0 < SDq; k0 += 4) {
            v2f a, bf;
            a[0]  = Abuf[lo * 65 + k0 + 2 * hi + 0];
            a[1]  = Abuf[lo * 65 + k0 + 2 * hi + 1];
            bf[0] = pw[(k0 + 2 * hi + 0) * SDq + n0 + lo];
            bf[1] = pw[(k0 + 2 * hi + 1) * SDq + n0 + lo];
            acc = wmma4(a, bf, acc);
        }
#pragma unroll
        for (int j = 0; j < 8; ++j) {
            int r = j + 8 * hi;
            H1[r * 65 + n0 + lo] =
                acc[j] + h0[((size_t)(row0 + r)) * SDq + n0 + lo];
        }
    }
    __syncthreads();

    // ---- phase 3: Y2 = LN2(H1) into Abuf (lane owns row lo, half hi)
    {
        float s = 0.f, s2 = 0.f;
#pragma unroll 4
        for (int c = 0; c < 32; ++c) {
            float v = H1[lo * 65 + hi * 32 + c];
            s += v; s2 += v * v;
        }
        s  += __shfl_xor(s, 16);
        s2 += __shfl_xor(s2, 16);
        float mean = s * (1.f / 64.f);
        float var  = s2 * (1.f / 64.f) - mean * mean;
        float rstd = rsqrtf(var + EPSq);
#pragma unroll 4
        for (int c = 0; c < 32; ++c) {
            int cc = hi * 32 + c;
            float v = H1[lo * 65 + cc];
            Abuf[lo * 65 + cc] = (v - mean) * rstd * g2[cc] + b2v[cc];
        }
    }
    __syncthreads();

    // ---- phase 4: Hid = relu(Y2 @ w1 + b1)   (WMMA, N=256)
    for (int n0 = 0; n0 < FHq; n0 += 16) {
        float bias = bb1[n0 + lo];
        v8f acc;
#pragma unroll
        for (int j = 0; j < 8; ++j) acc[j] = bias;
#pragma unroll
        for (int k0 = 0; k0 < SDq; k0 += 4) {
            v2f a, bf;
            a[0]  = Abuf[lo * 65 + k0 + 2 * hi + 0];
            a[1]  = Abuf[lo * 65 + k0 + 2 * hi + 1];
            bf[0] = w1[(k0 + 2 * hi + 0) * FHq + n0 + lo];
            bf[1] = w1[(k0 + 2 * hi + 1) * FHq + n0 + lo];
            acc = wmma4(a, bf, acc);
        }
#pragma unroll
        for (int j = 0; j < 8; ++j)
            Hid[(j + 8 * hi) * 257 + n0 + lo] = fmaxf(acc[j], 0.f);
    }
    __syncthreads();

    // ---- phase 5: hout = H1 + Hid @ w2 + b2   (WMMA, K=256)
    for (int n0 = 0; n0 < SDq; n0 += 16) {
        float bias = bb2[n0 + lo];
        v8f acc;
#pragma unroll
        for (int j = 0; j < 8; ++j) acc[j] = bias;
#pragma unroll 8
        for (int k0 = 0; k0 < FHq; k0 += 4) {
            v2f a, bf;
            a[0]  = Hid[lo * 257 + k0 + 2 * hi + 0];
            a[1]  = Hid[lo * 257 + k0 + 2 * hi + 1];
            bf[0] = w2[(k0 + 2 * hi + 0) * SDq + n0 + lo];
            bf[1] = w2[(k0 + 2 * hi + 1) * SDq + n0 + lo];
            acc = wmma4(a, bf, acc);
        }
#pragma unroll
        for (int j = 0; j < 8; ++j) {
            int r = j + 8 * hi;
            hout[((size_t)(row0 + r)) * SDq + n0 + lo] =
                acc[j] + H1[r * 65 + n0 + lo];
        }
    }
}

// =====================================================================
// K5: Booster  out[b,i,j,t] = bw[i,j,t] * h[b, j, t]   (134 MB stream-out)
// =====================================================================
__global__ __launch_bounds__(256) void k_booster(
    const float* __restrict__ h, const float* __restrict__ bw,
    float* __restrict__ out)
{
    unsigned id = blockIdx.x * 256u + threadIdx.x;   // < 33,554,432
    int t = id & (Tq - 1);
    unsigned r = id >> 11;
    int j = r & 63; r >>= 6;
    int i = r & 63; int b = (int)(r >> 6);
    out[id] = bw[((size_t)(i * SDq + j)) * Tq + t] *
              h[(size_t)b * (SDq * Tq) + (size_t)j * Tq + t];
}

// =====================================================================
extern "C" void kernel_launch(void* const* d_in, const int* in_sizes, int n_in,
                              void* d_out, int out_size, void* d_ws, size_t ws_size,
                              hipStream_t stream) {
    const float* x    = (const float*)d_in[0];
    const float* fw   = (const float*)d_in[1];
    const float* bw   = (const float*)d_in[2];
    const float* wq   = (const float*)d_in[3];
    const float* wk   = (const float*)d_in[4];
    const float* wvw  = (const float*)d_in[5];
    const float* pw   = (const float*)d_in[6];
    const float* pb   = (const float*)d_in[7];
    const float* g1   = (const float*)d_in[8];
    const float* b1v  = (const float*)d_in[9];
    const float* g2   = (const float*)d_in[10];
    const float* b2v  = (const float*)d_in[11];
    const float* w1   = (const float*)d_in[12];
    const float* bb1  = (const float*)d_in[13];
    const float* w2   = (const float*)d_in[14];
    const float* bb2  = (const float*)d_in[15];
    float* out = (float*)d_out;

    const size_t RT = (size_t)Bq * Tq * SDq;   // 524288 rows*64 elems
    float* W  = (float*)d_ws;
    float* h0 = W;                 // 2 MB
    float* q  = W + RT;            // 2 MB
    float* ck = W + 2 * RT;        // 256
    float* cv = ck + 256;          // 256
    float* mM = ck + 512;          // 256
    float* lL = ck + 768;          // 256
    float* h  = ck + 1024;         // 2 MB

    hipMemsetAsync(ck, 0, 512 * sizeof(float), stream);  // zero ck+cv

    k_feebler      <<<RT / 256, 256, 0, stream>>>(x, fw, h0);
    k_ln1_qkv      <<<(Bq * Tq) / 8, 256, 0, stream>>>(h0, g1, b1v, wq, wk, wvw,
                                                       q, ck, cv);
    k_softmax_stats<<<Bq * SDq, 256, 0, stream>>>(q, ck, mM, lL);
    k_attn_ffn     <<<(Bq * Tq) / 16, 32, 0, stream>>>(q, h0, ck, cv, mM, lL,
                                                       pw, pb, g2, b2v,
                                                       w1, bb1, w2, bb2, h);
    k_booster      <<<((size_t)Bq * Tq * NEq) / 256, 256, 0, stream>>>(h, bw, out);
}